// ARLoss_88390426951926
// MI455X (gfx1250) — compile-verified
//
#include <hip/hip_runtime.h>

typedef float f4  __attribute__((ext_vector_type(4)));
typedef float v2f __attribute__((ext_vector_type(2)));
typedef float v8f __attribute__((ext_vector_type(8)));

#define EPS 1e-6f

// ---- per-element loss (matches reference math; rintf = round-half-even) ----
__device__ __forceinline__ float loss_elem(float x, float t) {
    float c     = __builtin_rintf(t);          // classes = round(target), RNE
    float delta = x - c;
    bool  pos   = (delta >= 0.0f);
    float z     = pos ? __builtin_ceilf(x) : __builtin_floorf(x);
    float margin = __builtin_fabsf(delta) - __builtin_fabsf(x - z)
                 + (pos ? EPS : 0.0f);
    return fmaxf(margin, 0.0f);
}

// ---- exact wave32 sum via V_WMMA_F32_16X16X4_F32 ----
// A(16x4) holds each lane's value once (second A register = 0), B(4x16) = ones,
// so D[m][n] = sum_k A[m][k]. Lanes 0-15 hold D rows 0..7 across the 8 D regs,
// lanes 16-31 hold rows 8..15; summing the 8 regs per lane and one xor-16
// shuffle gives the full 32-lane sum in every lane. EXEC must be all ones:
// callers keep the whole block live (no early returns).
__device__ __forceinline__ float wave_reduce32(float v) {
    v2f a; a[0] = v;    a[1] = 0.0f;
    v2f b; b[0] = 1.0f; b[1] = 1.0f;
    v8f c = {};
    v8f d = __builtin_amdgcn_wmma_f32_16x16x4_f32(
        /*neg_a=*/false, a, /*neg_b=*/false, b,
        /*c_mod=*/(short)0, c, /*reuse_a=*/false, /*reuse_b=*/false);
    float t = ((d[0] + d[1]) + (d[2] + d[3])) + ((d[4] + d[5]) + (d[6] + d[7]));
    t += __shfl_xor(t, 16, 32);
    return t;
}

// ---- pass 1: streaming loss + per-block partial sums (HBM-bandwidth bound) ----
__global__ void __launch_bounds__(256)
arloss_partial(const float* __restrict__ x, const float* __restrict__ t,
               float* __restrict__ partial, int n4, int n) {
    __shared__ float smem[8];
    const int tid = threadIdx.x;
    const f4* __restrict__ x4 = (const f4*)x;
    const f4* __restrict__ t4 = (const f4*)t;

    float acc = 0.0f;
    int stride = gridDim.x * blockDim.x;
    for (int i = blockIdx.x * blockDim.x + tid; i < n4; i += stride) {
        // non-temporal 128-bit loads: 256MB streamed once, don't thrash L2
        f4 xv = __builtin_nontemporal_load(&x4[i]);
        f4 tv = __builtin_nontemporal_load(&t4[i]);
        acc += loss_elem(xv[0], tv[0]);
        acc += loss_elem(xv[1], tv[1]);
        acc += loss_elem(xv[2], tv[2]);
        acc += loss_elem(xv[3], tv[3]);
    }
    // tail (N % 4), handled by a single thread; wave re-converges afterwards
    if (blockIdx.x == 0 && tid == 0) {
        for (int i = n4 * 4; i < n; ++i) acc += loss_elem(x[i], t[i]);
    }

    float w = wave_reduce32(acc);          // full EXEC: every thread alive
    if ((tid & 31) == 0) smem[tid >> 5] = w;
    __syncthreads();
    if (tid == 0) {
        float s = smem[0];
        #pragma unroll
        for (int i = 1; i < 8; ++i) s += smem[i];   // fixed order: deterministic
        partial[blockIdx.x] = s;
    }
}

// ---- pass 2: deterministic reduction of block partials, scale by 1/N ----
__global__ void __launch_bounds__(256)
arloss_final(const float* __restrict__ partial, float* __restrict__ out,
             int nparts, float inv_n) {
    __shared__ float smem[8];
    const int tid = threadIdx.x;
    float acc = 0.0f;
    for (int i = tid; i < nparts; i += 256) acc += partial[i];

    float w = wave_reduce32(acc);
    if ((tid & 31) == 0) smem[tid >> 5] = w;
    __syncthreads();
    if (tid == 0) {
        float s = smem[0];
        #pragma unroll
        for (int i = 1; i < 8; ++i) s += smem[i];
        out[0] = s * inv_n;
    }
}

extern "C" void kernel_launch(void* const* d_in, const int* in_sizes, int n_in,
                              void* d_out, int out_size, void* d_ws, size_t ws_size,
                              hipStream_t stream) {
    const float* x = (const float*)d_in[0];   // input  (8192*4096 f32)
    const float* t = (const float*)d_in[1];   // target (8192*4096 f32)
    float* out = (float*)d_out;               // scalar f32
    float* ws  = (float*)d_ws;                // block partials

    const int n  = in_sizes[0];
    const int n4 = n / 4;
    const int blocks = 2048;                  // 16K wave32s; 16 f4-iters/thread

    arloss_partial<<<blocks, 256, 0, stream>>>(x, t, ws, n4, n);
    arloss_final<<<1, 256, 0, stream>>>(ws, out, blocks, 1.0f / (float)n);
}